// SMPL_Layer_61692910240326
// MI455X (gfx1250) — compile-verified
//
#include <hip/hip_runtime.h>
#include <hip/hip_bf16.h>

#define SMPL_V 6890
#define SMPL_B 1024
#define SMPL_J 24
#define SMPL_NB 10
#define KF 220   // padded fused feature length: 207 pose | 0 | 10 betas | 1.0 | 0

typedef __attribute__((ext_vector_type(2))) float v2f;
typedef __attribute__((ext_vector_type(8))) float v8f;

__constant__ int c_parents[SMPL_J] = {-1,0,0,0,1,2,3,4,5,6,7,8,9,9,9,12,13,14,16,17,18,19,20,21};

__device__ __forceinline__ v8f wmma4_f32(v2f a, v2f b, v8f c) {
  // V_WMMA_F32_16X16X4_F32 : D(16x16 f32) = A(16x4 f32) * B(4x16 f32) + C
  return __builtin_amdgcn_wmma_f32_16x16x4_f32(false, a, false, b, (short)0, c, false, false);
}

// ---------------------------------------------------------------------------
// Kernel A0: precompute Jt0 = Jreg @ v_template  (24x3)
//            and JS[(j*3+c)*10+k] = sum_v Jreg[j,v]*shapedirs[(v*3+c),k]
// ---------------------------------------------------------------------------
__global__ __launch_bounds__(128) void smpl_jreg_kernel(
    const float* __restrict__ Jreg, const float* __restrict__ vtmpl,
    const float* __restrict__ shapedirs, float* __restrict__ Jt0JS) {
  int t = blockIdx.x * blockDim.x + threadIdx.x;
  if (t >= SMPL_J * 3) return;
  int j = t / 3, c = t % 3;
  float acc0 = 0.f;
  float accK[SMPL_NB];
  #pragma unroll
  for (int k = 0; k < SMPL_NB; ++k) accK[k] = 0.f;
  for (int v = 0; v < SMPL_V; ++v) {
    float w = Jreg[(size_t)j * SMPL_V + v];
    acc0 += w * vtmpl[v * 3 + c];
    const float* sd = shapedirs + (size_t)(v * 3 + c) * SMPL_NB;
    #pragma unroll
    for (int k = 0; k < SMPL_NB; ++k) accK[k] += w * sd[k];
  }
  Jt0JS[j * 3 + c] = acc0;
  #pragma unroll
  for (int k = 0; k < SMPL_NB; ++k) Jt0JS[72 + (j * 3 + c) * SMPL_NB + k] = accK[k];
}

// ---------------------------------------------------------------------------
// Kernel A: per-batch Rodrigues, fused feature row F, kinematic chain,
//           A_skin (24 x [3x4]) and jtr output.
// ---------------------------------------------------------------------------
__global__ __launch_bounds__(128) void smpl_batch_kernel(
    const float* __restrict__ pose, const float* __restrict__ betas,
    const float* __restrict__ trans, const float* __restrict__ Jt0JS,
    float* __restrict__ F, float* __restrict__ As, float* __restrict__ jtr_out) {
  int b = blockIdx.x * blockDim.x + threadIdx.x;
  if (b >= SMPL_B) return;

  float R[SMPL_J][9];
  #pragma unroll 1
  for (int j = 0; j < SMPL_J; ++j) {
    float ax = pose[b * 72 + j * 3 + 0];
    float ay = pose[b * 72 + j * 3 + 1];
    float az = pose[b * 72 + j * 3 + 2];
    float ang = sqrtf(ax * ax + ay * ay + az * az + 1e-12f);
    float inv = 1.f / ang;
    float rx = ax * inv, ry = ay * inv, rz = az * inv;
    float cc = cosf(ang), ss = sinf(ang), tt = 1.f - cc;
    R[j][0] = cc + tt * rx * rx;      R[j][1] = tt * rx * ry - ss * rz; R[j][2] = tt * rx * rz + ss * ry;
    R[j][3] = tt * rx * ry + ss * rz; R[j][4] = cc + tt * ry * ry;      R[j][5] = tt * ry * rz - ss * rx;
    R[j][6] = tt * rx * rz - ss * ry; R[j][7] = tt * ry * rz + ss * rx; R[j][8] = cc + tt * rz * rz;
  }

  // fused feature row: [pose_feat(207), 0, betas(10), 1, 0]
  float* Fr = F + (size_t)b * KF;
  for (int j = 1; j < SMPL_J; ++j) {
    #pragma unroll
    for (int e = 0; e < 9; ++e)
      Fr[(j - 1) * 9 + e] = R[j][e] - ((e == 0 || e == 4 || e == 8) ? 1.f : 0.f);
  }
  Fr[207] = 0.f;
  #pragma unroll
  for (int k = 0; k < SMPL_NB; ++k) Fr[208 + k] = betas[b * SMPL_NB + k];
  Fr[218] = 1.f;
  Fr[219] = 0.f;

  // joint positions from precomputed regression
  float Jts[SMPL_J][3];
  const float* Jt0 = Jt0JS;
  const float* JS = Jt0JS + 72;
  for (int j = 0; j < SMPL_J; ++j) {
    #pragma unroll
    for (int c = 0; c < 3; ++c) {
      float acc = Jt0[j * 3 + c];
      #pragma unroll
      for (int k = 0; k < SMPL_NB; ++k) acc += betas[b * SMPL_NB + k] * JS[(j * 3 + c) * SMPL_NB + k];
      Jts[j][c] = acc;
    }
  }

  // kinematic chain, 3x4 transforms (implicit [0,0,0,1] bottom row)
  float Tm[SMPL_J][12];
  #pragma unroll
  for (int r = 0; r < 3; ++r) {
    Tm[0][r * 4 + 0] = R[0][r * 3 + 0];
    Tm[0][r * 4 + 1] = R[0][r * 3 + 1];
    Tm[0][r * 4 + 2] = R[0][r * 3 + 2];
    Tm[0][r * 4 + 3] = Jts[0][r];
  }
  for (int j = 1; j < SMPL_J; ++j) {
    int p = c_parents[j];
    float L[12];
    #pragma unroll
    for (int r = 0; r < 3; ++r) {
      L[r * 4 + 0] = R[j][r * 3 + 0];
      L[r * 4 + 1] = R[j][r * 3 + 1];
      L[r * 4 + 2] = R[j][r * 3 + 2];
      L[r * 4 + 3] = Jts[j][r] - Jts[p][r];
    }
    #pragma unroll
    for (int r = 0; r < 3; ++r) {
      #pragma unroll
      for (int c = 0; c < 4; ++c) {
        float acc = (c == 3) ? Tm[p][r * 4 + 3] : 0.f;
        #pragma unroll
        for (int k = 0; k < 3; ++k) acc += Tm[p][r * 4 + k] * L[k * 4 + c];
        Tm[j][r * 4 + c] = acc;
      }
    }
  }

  float tx = trans[b * 3 + 0], ty = trans[b * 3 + 1], tz = trans[b * 3 + 2];
  for (int j = 0; j < SMPL_J; ++j) {
    float jx = Tm[j][3], jy = Tm[j][7], jz = Tm[j][11];
    jtr_out[((size_t)b * SMPL_J + j) * 3 + 0] = jx + tx;
    jtr_out[((size_t)b * SMPL_J + j) * 3 + 1] = jy + ty;
    jtr_out[((size_t)b * SMPL_J + j) * 3 + 2] = jz + tz;
    float* dst = As + ((size_t)b * SMPL_J + j) * 12;
    #pragma unroll
    for (int r = 0; r < 3; ++r) {
      float tc = Tm[j][r * 4 + 3] - (Tm[j][r * 4 + 0] * Jts[j][0] +
                                     Tm[j][r * 4 + 1] * Jts[j][1] +
                                     Tm[j][r * 4 + 2] * Jts[j][2]);
      dst[r * 4 + 0] = Tm[j][r * 4 + 0];
      dst[r * 4 + 1] = Tm[j][r * 4 + 1];
      dst[r * 4 + 2] = Tm[j][r * 4 + 2];
      dst[r * 4 + 3] = tc;
    }
  }
}

// ---------------------------------------------------------------------------
// LBS skinning + store for one 16-batch accumulator set.
// C-tile layout: VGPR r -> batch bbase + r + 8*hf, lane&15 -> vertex column.
// ---------------------------------------------------------------------------
__device__ __forceinline__ void skin_store(
    const float* __restrict__ Aslds, const float* __restrict__ Wlds,
    const float* __restrict__ trans, float* __restrict__ verts,
    v8f cx, v8f cy, v8f cz, int bbase, int hf, int n, int wave, int v) {
  #pragma unroll 1
  for (int r = 0; r < 8; ++r) {
    const int bl = r + 8 * hf;
    const int b = bbase + bl;
    const float x = cx[r], y = cy[r], z = cz[r];
    float T[12];
    #pragma unroll
    for (int e = 0; e < 12; ++e) T[e] = 0.f;
    for (int j = 0; j < SMPL_J; ++j) {
      const float w = Wlds[(wave * 16 + n) * SMPL_J + j];
      const float* Aj = &Aslds[(bl * SMPL_J + j) * 12];
      #pragma unroll
      for (int e = 0; e < 12; ++e) T[e] += w * Aj[e];
    }
    if (v < SMPL_V) {
      #pragma unroll
      for (int mrow = 0; mrow < 3; ++mrow) {
        float val = T[4 * mrow + 0] * x + T[4 * mrow + 1] * y +
                    T[4 * mrow + 2] * z + T[4 * mrow + 3] + trans[b * 3 + mrow];
        verts[((size_t)b * SMPL_V + v) * 3 + mrow] = val;
      }
    }
  }
}

// ---------------------------------------------------------------------------
// Kernel B: fused blendshape GEMM (WMMA f32 16x16x4) + LBS skinning.
// Block: 256 threads = 8 waves, covers 32 batches x 128 vertices.
// Each wave owns two 16x16 output tiles per component (batch tiles 0 and 1)
// sharing every B-operand fetch -> inner loop is 6 wmma : 3 global b64 loads.
// ---------------------------------------------------------------------------
__global__ __launch_bounds__(256) void smpl_verts_kernel(
    const float* __restrict__ F, const float* __restrict__ As,
    const float* __restrict__ posedirs, const float* __restrict__ shapedirs,
    const float* __restrict__ vtmpl, const float* __restrict__ weights,
    const float* __restrict__ trans, float* __restrict__ verts) {
  __shared__ float Flds[32 * KF];           // feature tile (32 batches), 28.2 KB
  __shared__ float Wlds[128 * SMPL_J];      // skinning weights, 12.3 KB
  __shared__ float Aslds[16 * SMPL_J * 12]; // A_skin tile (phased), 18.4 KB

  const int tid = threadIdx.x;
  const int lane = tid & 31;
  const int wave = tid >> 5;
  const int b0 = blockIdx.y * 32;
  const int vblock = blockIdx.x * 128;

  for (int i = tid; i < 32 * KF; i += 256) {
    int bb = i / KF, kk = i % KF;
    Flds[i] = F[(size_t)(b0 + bb) * KF + kk];
  }
  for (int i = tid; i < 128 * SMPL_J; i += 256) {
    int vv = vblock + i / SMPL_J;
    int jj = i % SMPL_J;
    int vsafe = vv < SMPL_V ? vv : SMPL_V - 1;
    Wlds[i] = weights[(size_t)vsafe * SMPL_J + jj];
  }
  __syncthreads();

  const int hf = lane >> 4;   // K-half selector for f32 WMMA operand layout
  const int n = lane & 15;    // A: batch row (M); B: vertex column (N)
  const int v = vblock + wave * 16 + n;
  const int vs = v < SMPL_V ? v : SMPL_V - 1;
  const size_t r0 = (size_t)(vs * 3 + 0) * 207;
  const size_t r1 = (size_t)(vs * 3 + 1) * 207;
  const size_t r2 = (size_t)(vs * 3 + 2) * 207;

  v8f c00 = {}, c01 = {}, c02 = {};   // batch tile 0 (b0 .. b0+15)
  v8f c10 = {}, c11 = {}, c12 = {};   // batch tile 1 (b0+16 .. b0+31)

  // Main pose-blendshape steps: K indices 0..203, pure posedirs.
  for (int step = 0; step < 51; ++step) {
    const int kk = step * 4 + 2 * hf;
    v2f a0, a1;
    a0.x = Flds[n * KF + kk];
    a0.y = Flds[n * KF + kk + 1];
    a1.x = Flds[(n + 16) * KF + kk];
    a1.y = Flds[(n + 16) * KF + kk + 1];
    v2f bv;
    bv.x = posedirs[r0 + kk]; bv.y = posedirs[r0 + kk + 1];
    c00 = wmma4_f32(a0, bv, c00);
    c10 = wmma4_f32(a1, bv, c10);
    bv.x = posedirs[r1 + kk]; bv.y = posedirs[r1 + kk + 1];
    c01 = wmma4_f32(a0, bv, c01);
    c11 = wmma4_f32(a1, bv, c11);
    bv.x = posedirs[r2 + kk]; bv.y = posedirs[r2 + kk + 1];
    c02 = wmma4_f32(a0, bv, c02);
    c12 = wmma4_f32(a1, bv, c12);
  }
  // Tail pose step: K indices 204..207 (207 = zero pad). Branch-free masking.
  {
    const int kk = 204 + 2 * hf;
    const int ky = (kk + 1 < 207) ? (kk + 1) : 206;  // clamped index
    const float my = (kk + 1 < 207) ? 1.f : 0.f;     // zero mask for k==207
    v2f a0, a1;
    a0.x = Flds[n * KF + kk];
    a0.y = Flds[n * KF + kk + 1];   // Flds[..207] == 0
    a1.x = Flds[(n + 16) * KF + kk];
    a1.y = Flds[(n + 16) * KF + kk + 1];
    v2f bv;
    bv.x = posedirs[r0 + kk]; bv.y = posedirs[r0 + ky] * my;
    c00 = wmma4_f32(a0, bv, c00);
    c10 = wmma4_f32(a1, bv, c10);
    bv.x = posedirs[r1 + kk]; bv.y = posedirs[r1 + ky] * my;
    c01 = wmma4_f32(a0, bv, c01);
    c11 = wmma4_f32(a1, bv, c11);
    bv.x = posedirs[r2 + kk]; bv.y = posedirs[r2 + ky] * my;
    c02 = wmma4_f32(a0, bv, c02);
    c12 = wmma4_f32(a1, bv, c12);
  }
  // Shape-blendshape + template steps: K indices 208..219.
  //   208..217 -> shapedirs columns, 218 -> v_template (coef 1.0), 219 -> pad 0.
  // Branch-free: clamped loads of both candidates + selects.
  for (int step = 0; step < 3; ++step) {
    const int kk = 208 + step * 4 + 2 * hf;
    const int sk = kk - 208;                 // in {0,2,4,6,8,10}
    const int skx = sk < SMPL_NB ? sk : SMPL_NB - 1;
    const int sky = (sk + 1) < SMPL_NB ? (sk + 1) : SMPL_NB - 1;
    v2f a0, a1;
    a0.x = Flds[n * KF + kk];
    a0.y = Flds[n * KF + kk + 1];
    a1.x = Flds[(n + 16) * KF + kk];
    a1.y = Flds[(n + 16) * KF + kk + 1];
    #pragma unroll
    for (int c = 0; c < 3; ++c) {
      const int vc = vs * 3 + c;
      const float tv = vtmpl[vc];
      const float sx = shapedirs[(size_t)vc * SMPL_NB + skx];
      const float sy = shapedirs[(size_t)vc * SMPL_NB + sky];
      v2f bv;
      bv.x = (sk < SMPL_NB) ? sx : ((sk == SMPL_NB) ? tv : 0.f);
      bv.y = (sk + 1 < SMPL_NB) ? sy : ((sk + 1 == SMPL_NB) ? tv : 0.f);
      if (c == 0) { c00 = wmma4_f32(a0, bv, c00); c10 = wmma4_f32(a1, bv, c10); }
      else if (c == 1) { c01 = wmma4_f32(a0, bv, c01); c11 = wmma4_f32(a1, bv, c11); }
      else { c02 = wmma4_f32(a0, bv, c02); c12 = wmma4_f32(a1, bv, c12); }
    }
  }

  // ---- Skinning phase 0: batches b0 .. b0+15 ----
  __syncthreads();
  for (int i = tid; i < 16 * SMPL_J * 12; i += 256) {
    int bb = i / (SMPL_J * 12), e = i % (SMPL_J * 12);
    Aslds[i] = As[(size_t)(b0 + bb) * (SMPL_J * 12) + e];
  }
  __syncthreads();
  skin_store(Aslds, Wlds, trans, verts, c00, c01, c02, b0, hf, n, wave, v);

  // ---- Skinning phase 1: batches b0+16 .. b0+31 ----
  __syncthreads();
  for (int i = tid; i < 16 * SMPL_J * 12; i += 256) {
    int bb = i / (SMPL_J * 12), e = i % (SMPL_J * 12);
    Aslds[i] = As[(size_t)(b0 + 16 + bb) * (SMPL_J * 12) + e];
  }
  __syncthreads();
  skin_store(Aslds, Wlds, trans, verts, c10, c11, c12, b0 + 16, hf, n, wave, v);
}

// ---------------------------------------------------------------------------
extern "C" void kernel_launch(void* const* d_in, const int* in_sizes, int n_in,
                              void* d_out, int out_size, void* d_ws, size_t ws_size,
                              hipStream_t stream) {
  (void)in_sizes; (void)n_in; (void)out_size; (void)ws_size;
  const float* pose      = (const float*)d_in[0];
  const float* betas     = (const float*)d_in[1];
  const float* trans     = (const float*)d_in[2];
  const float* vtmpl     = (const float*)d_in[3];
  const float* shapedirs = (const float*)d_in[4];
  const float* posedirs  = (const float*)d_in[5];
  const float* Jreg      = (const float*)d_in[6];
  const float* weights   = (const float*)d_in[7];

  float* out_verts = (float*)d_out;
  float* out_jtr   = (float*)d_out + (size_t)SMPL_B * SMPL_V * 3;

  // Workspace layout (~2.1 MB total):
  const size_t F_bytes  = (size_t)SMPL_B * KF * 4;                 // 901,120
  const size_t F_off    = 0;
  const size_t As_off   = (F_bytes + 255) & ~(size_t)255;          // 901,120
  const size_t As_bytes = (size_t)SMPL_B * SMPL_J * 12 * 4;        // 1,179,648
  const size_t J_off    = As_off + As_bytes;                       // 2,080,768

  float* F     = (float*)((char*)d_ws + F_off);
  float* As    = (float*)((char*)d_ws + As_off);
  float* Jt0JS = (float*)((char*)d_ws + J_off);

  smpl_jreg_kernel<<<1, 128, 0, stream>>>(Jreg, vtmpl, shapedirs, Jt0JS);
  smpl_batch_kernel<<<SMPL_B / 128, 128, 0, stream>>>(pose, betas, trans, Jt0JS,
                                                      F, As, out_jtr);
  dim3 gridB((SMPL_V + 127) / 128, SMPL_B / 32);
  smpl_verts_kernel<<<gridB, 256, 0, stream>>>(F, As, posedirs, shapedirs, vtmpl,
                                               weights, trans, out_verts);
}